// OutputParser_20169166422203
// MI455X (gfx1250) — compile-verified
//
#include <hip/hip_runtime.h>
#include <math.h>

#define NN 8
#define BB 1000
#define CC 80
#define KK 100
#define IMG_H 800.0f
#define IMG_W 800.0f
#define NEGV (-1.0e10f)
#define SCORE_TH 0.05f
#define IOU_TH 0.5f

// ---------- CDNA5 async global->LDS staging helpers ----------
typedef __attribute__((address_space(3))) void lds_void;

__device__ __forceinline__ unsigned lds_off(void* p) {
  // generic -> LDS addrspace cast strips the flat aperture; AS(3) ptrs are 32-bit
  return (unsigned)(unsigned long long)(lds_void*)p;
}

// GLOBAL_LOAD_ASYNC_TO_LDS_B128, GV mode (64-bit vaddr). Tracked by ASYNCcnt.
__device__ __forceinline__ void async_g2l_b128(void* lds_dst, const void* gsrc) {
  unsigned loff = lds_off(lds_dst);
  unsigned long long ga = (unsigned long long)gsrc;
  asm volatile("global_load_async_to_lds_b128 %0, %1, off"
               :: "v"(loff), "v"(ga) : "memory");
}

__device__ __forceinline__ void wait_async0() {
  asm volatile("s_wait_asynccnt 0" ::: "memory");
}

// ---------- Kernel 1: decode boxes into (N,C,B,4) + scores (N,C,B) ----------
// 3D grid: x covers B, y = class, z = image  -> zero integer divisions.
__global__ __launch_bounds__(128) void decode_kernel(
    const float* __restrict__ rois,      // (N,B,4)  y1,x1,y2,x2
    const float* __restrict__ conf,      // (N,B,C+1)
    const float* __restrict__ deltas,    // (N,B,C*4)
    float* __restrict__ boxes_nc,        // (N,C,B,4)
    float* __restrict__ scores_nc)       // (N,C,B)
{
  const int b = blockIdx.x * 128 + threadIdx.x;
  if (b >= BB) return;
  const int c = blockIdx.y;
  const int n = blockIdx.z;

  float4 r = ((const float4*)rois)[n * BB + b];          // (y1,x1,y2,x2)
  float4 d = ((const float4*)deltas)[(size_t)(n * BB + b) * CC + c];

  float w0 = r.w - r.y + 1.0f;
  float h0 = r.z - r.x + 1.0f;
  float x0 = r.y + 0.5f * w0;
  float y0 = r.x + 0.5f * h0;

  float cx = (d.x * 0.1f) * w0 + x0;
  float cy = (d.y * 0.1f) * h0 + y0;
  float ww = expf(d.z * 0.2f) * w0;
  float hh = expf(d.w * 0.2f) * h0;

  float xx1 = fminf(fmaxf(cx - 0.5f * ww, 0.0f), IMG_W - 1.0f);
  float yy1 = fminf(fmaxf(cy - 0.5f * hh, 0.0f), IMG_H - 1.0f);
  float xx2 = fminf(fmaxf(cx + 0.5f * ww, 0.0f), IMG_W - 1.0f);
  float yy2 = fminf(fmaxf(cy + 0.5f * hh, 0.0f), IMG_H - 1.0f);

  size_t o = (size_t)(n * CC + c) * BB + b;
  ((float4*)boxes_nc)[o] = make_float4(yy1, xx1, yy2, xx2);
  scores_nc[o] = conf[(size_t)(n * BB + b) * (CC + 1) + c];
}

// ---------- Kernel 2: greedy NMS, ONE WAVE32 per (n,c) problem ----------
// No cross-wave barriers: argmax = strided scan + shfl reduce + shfl broadcast.
__global__ __launch_bounds__(32) void nms_kernel(
    const float* __restrict__ boxes_nc, const float* __restrict__ scores_nc,
    float* __restrict__ flat_score,   // (N, C*K): -1 if invalid
    float* __restrict__ flat_box,     // (N, C*K, 4)
    float* __restrict__ flat_valid)   // (N, C*K): 1/0
{
  __shared__ float4 sbox[BB];          // 16000 B
  __shared__ float  sscore[BB];        //  4000 B
  __shared__ float  sarea[BB];         //  4000 B
  __shared__ float  swork[BB];         //  4000 B   -> 28 KB/block, ~11 blocks/WGP

  const int nc   = blockIdx.x;         // n*CC + c
  const int lane = threadIdx.x;        // 0..31
  const float4* gbox = (const float4*)boxes_nc + (size_t)nc * BB;
  const float*  gsc  = scores_nc + (size_t)nc * BB;

  // Async-DMA the full candidate set into LDS (boxes: 1000x16B, scores: 250x16B)
  for (int t = lane; t < BB; t += 32)
    async_g2l_b128(&sbox[t], &gbox[t]);
  for (int t = lane; t < BB / 4; t += 32)
    async_g2l_b128(&((float4*)sscore)[t], &((const float4*)gsc)[t]);
  wait_async0();
  __syncthreads();   // single-wave WG: lowers to cheap fence, orders LDS visibility

  for (int t = lane; t < BB; t += 32) {
    float4 bx = sbox[t];
    sarea[t] = (bx.z - bx.x) * (bx.w - bx.y);
    float s = sscore[t];
    swork[t] = (s > SCORE_TH) ? s : NEGV;
  }
  __syncthreads();

  for (int it = 0; it < KK; ++it) {
    // ---- wave argmax over swork (prefer lower index on ties) ----
    float v = -1.0e38f; int idx = 0;
    for (int t = lane; t < BB; t += 32) {
      float w = swork[t];
      if (w > v) { v = w; idx = t; }
    }
    for (int off = 16; off > 0; off >>= 1) {
      float ov = __shfl_down(v, off);
      int   oi = __shfl_down(idx, off);
      if (ov > v || (ov == v && oi < idx)) { v = ov; idx = oi; }
    }
    const int   best  = __shfl(idx, 0);
    const float bestv = __shfl(v, 0);
    const bool  valid = bestv > (NEGV * 0.5f);
    const float4 bb = sbox[best];      // same-address LDS broadcast
    const float  ab = sarea[best];

    // ---- suppress ----
    for (int t = lane; t < BB; t += 32) {
      float4 ob = sbox[t];
      float iy1 = fmaxf(bb.x, ob.x), ix1 = fmaxf(bb.y, ob.y);
      float iy2 = fminf(bb.z, ob.z), ix2 = fminf(bb.w, ob.w);
      float inter = fmaxf(iy2 - iy1, 0.0f) * fmaxf(ix2 - ix1, 0.0f);
      float iou = inter / fmaxf(ab + sarea[t] - inter, 1e-8f);
      if (iou > IOU_TH || t == best) swork[t] = NEGV;
    }
    if (lane == 0) {
      size_t o = (size_t)nc * KK + it;
      flat_score[o] = valid ? sscore[best] : -1.0f;
      flat_valid[o] = valid ? 1.0f : 0.0f;
      ((float4*)flat_box)[o] = bb;
    }
    __syncthreads();   // order swork stores before next iteration's scan
  }
}

// ---------- Kernel 3: per-image top-100 over C*K=8000 candidates ----------
__global__ __launch_bounds__(256) void topk_kernel(
    const float* __restrict__ flat_score, const float* __restrict__ flat_box,
    const float* __restrict__ flat_valid, float* __restrict__ out)
{
  __shared__ float sval[CC * KK];      // 32 KB
  __shared__ float rv[8];
  __shared__ int   ri[8];
  __shared__ float bval;
  __shared__ int   bidx;
  __shared__ int   cnt;

  const int n = blockIdx.x, tid = threadIdx.x;
  const float* gs = flat_score + (size_t)n * CC * KK;

  for (int t = tid; t < (CC * KK) / 4; t += 256)
    async_g2l_b128(&((float4*)sval)[t], &((const float4*)gs)[t]);
  wait_async0();
  if (tid == 0) cnt = 0;
  __syncthreads();

  float* nb  = out;                     // N*K*4 = 3200
  float* ns  = out + NN * KK * 4;       // + 800
  float* ncl = ns + NN * KK;            // + 800
  float* nd  = ncl + NN * KK;           // + 8

  const int lane = tid & 31, wave = tid >> 5;

  for (int it = 0; it < KK; ++it) {
    float v = -1.0e38f; int idx = 0;
    for (int t = tid; t < CC * KK; t += 256) {
      float w = sval[t];
      if (w > v) { v = w; idx = t; }
    }
    for (int off = 16; off > 0; off >>= 1) {
      float ov = __shfl_down(v, off);
      int   oi = __shfl_down(idx, off);
      if (ov > v || (ov == v && oi < idx)) { v = ov; idx = oi; }
    }
    if (lane == 0) { rv[wave] = v; ri[wave] = idx; }
    __syncthreads();
    if (tid == 0) {
      float mv = rv[0]; int mi = ri[0];
      for (int w2 = 1; w2 < 8; ++w2)
        if (rv[w2] > mv || (rv[w2] == mv && ri[w2] < mi)) { mv = rv[w2]; mi = ri[w2]; }
      bool valid = flat_valid[(size_t)n * CC * KK + mi] != 0.0f;
      size_t o = (size_t)n * KK + it;
      float4 bx = valid ? ((const float4*)flat_box)[(size_t)n * CC * KK + mi]
                        : make_float4(0.f, 0.f, 0.f, 0.f);
      ((float4*)nb)[o] = bx;
      ns[o]  = valid ? mv : 0.0f;
      ncl[o] = valid ? (float)(mi / KK) : 0.0f;
      if (valid) cnt++;
      sval[mi] = -1.0e30f;              // remove from further consideration
    }
    __syncthreads();
  }
  if (tid == 0) nd[n] = (float)cnt;
}

// ---------- Kernel 4: rois pass-through ----------
__global__ __launch_bounds__(256) void copy_rois_kernel(
    const float* __restrict__ rois, float* __restrict__ out_rois)
{
  int i = blockIdx.x * blockDim.x + threadIdx.x;
  if (i < NN * BB * 4) out_rois[i] = rois[i];
}

extern "C" void kernel_launch(void* const* d_in, const int* in_sizes, int n_in,
                              void* d_out, int out_size, void* d_ws, size_t ws_size,
                              hipStream_t stream) {
  (void)in_sizes; (void)n_in; (void)out_size; (void)ws_size;
  const float* rois   = (const float*)d_in[0];   // (8,1000,4)
  const float* conf   = (const float*)d_in[1];   // (8,1000,81)
  const float* deltas = (const float*)d_in[2];   // (8,1000,320)
  // d_in[3] = input_image: only shape (800x800) is needed, hardcoded.

  float* ws = (float*)d_ws;
  // workspace layout (floats); total 3,584,000 floats = 13.7 MB
  float* boxes_nc   = ws;                               // 2,560,000
  float* scores_nc  = boxes_nc  + (size_t)NN*CC*BB*4;   //   640,000
  float* flat_score = scores_nc + (size_t)NN*CC*BB;     //    64,000
  float* flat_box   = flat_score + (size_t)NN*CC*KK;    //   256,000
  float* flat_valid = flat_box   + (size_t)NN*CC*KK*4;  //    64,000

  float* out = (float*)d_out;

  dim3 dgrid((BB + 127) / 128, CC, NN);
  decode_kernel<<<dgrid, 128, 0, stream>>>(rois, conf, deltas, boxes_nc, scores_nc);

  nms_kernel<<<NN * CC, 32, 0, stream>>>(
      boxes_nc, scores_nc, flat_score, flat_box, flat_valid);

  topk_kernel<<<NN, 256, 0, stream>>>(flat_score, flat_box, flat_valid, out);

  copy_rois_kernel<<<(NN * BB * 4 + 255) / 256, 256, 0, stream>>>(
      rois, out + NN * KK * 4 + NN * KK + NN * KK + NN);
}